// ForceModel_62989990363468
// MI455X (gfx1250) — compile-verified
//
#include <hip/hip_runtime.h>

#define NBATCH 4
#define NP     2048
#define FEPS   1e-12f

typedef float v2f __attribute__((ext_vector_type(2)));
typedef float v8f __attribute__((ext_vector_type(8)));

struct V3 { float x, y, z; };
__device__ __forceinline__ V3 v3sub(V3 a, V3 b) { return {a.x-b.x, a.y-b.y, a.z-b.z}; }
__device__ __forceinline__ V3 v3add(V3 a, V3 b) { return {a.x+b.x, a.y+b.y, a.z+b.z}; }
__device__ __forceinline__ V3 v3scale(V3 a, float s) { return {a.x*s, a.y*s, a.z*s}; }
__device__ __forceinline__ float v3dot(V3 a, V3 b) { return a.x*b.x + a.y*b.y + a.z*b.z; }
__device__ __forceinline__ V3 v3cross(V3 a, V3 b) {
    return {a.y*b.z - a.z*b.y, a.z*b.x - a.x*b.z, a.x*b.y - a.y*b.x};
}
__device__ __forceinline__ V3 loadp(const float* pos, int bn) {
    const float* p = pos + (size_t)bn * 3;
    return {p[0], p[1], p[2]};
}
__device__ __forceinline__ void atomicAdd3(float* out, int bn, V3 f) {
    float* o = out + (size_t)bn * 3;
    atomicAdd(o + 0, f.x);
    atomicAdd(o + 1, f.y);
    atomicAdd(o + 2, f.z);
}

// upper-triangular row start: rowbase(a) = a*N - a*(a+1)/2  (pure ALU, no load)
__device__ __forceinline__ int rowbase_of(int a) {
    return (a << 11) - ((a * (a + 1)) >> 1);
}

// -------------------- zero output --------------------
__global__ void __launch_bounds__(256) k_zero(float* out, int n) {
    int t = blockIdx.x * 256 + threadIdx.x;
    if (t < n) out[t] = 0.0f;
}

// ---------- prep: pack per-particle rows into workspace ----------
// row layout (8 floats): x, y, z, |p|^2, q, radius, 0, 0
__global__ void __launch_bounds__(256) k_prep(const float* __restrict__ pos,
                                              const float* __restrict__ q_vals,
                                              const int* __restrict__ charge_idx,
                                              const float* __restrict__ bead_radii,
                                              const int* __restrict__ bead_types,
                                              float* __restrict__ ws) {
    int t = blockIdx.x * 256 + threadIdx.x;
    if (t >= NBATCH * NP) return;
    int n = t & (NP - 1);
    const float* p = pos + (size_t)t * 3;
    float x = p[0], y = p[1], z = p[2];
    float* w = ws + (size_t)t * 8;
    w[0] = x; w[1] = y; w[2] = z;
    w[3] = x*x + y*y + z*z;
    w[4] = q_vals[charge_idx[n]];
    w[5] = bead_radii[bead_types[n]];
    w[6] = 0.0f;
    w[7] = 0.0f;
}

// -------------------- nonbonded all-pairs (WMMA tiled) --------------------
// Each wave owns one 16-particle j-tile of one batch and one of 4 i-slices.
// Gram matrix p_i . p_j via V_WMMA_F32_16X16X4_F32; force on j accumulated
// per-lane (lane&15 == column N), final half-wave reduction via shfl_xor(16).
__global__ void __launch_bounds__(256) k_nonbond(const float* __restrict__ ws,
                                                 const int* __restrict__ nb_tid,
                                                 const float* __restrict__ disp_k,
                                                 const float* __restrict__ hbond_k,
                                                 const float* __restrict__ f_0,
                                                 const float* __restrict__ e_r,
                                                 float* __restrict__ out) {
    __shared__ float sDisp[256];
    __shared__ float sHbond[256];
    sDisp[threadIdx.x]  = disp_k[threadIdx.x];
    sHbond[threadIdx.x] = hbond_k[threadIdx.x];
    __syncthreads();

    const float Cc = f_0[0] / e_r[0];

    int lane = threadIdx.x & 31;
    int wave = threadIdx.x >> 5;
    int w = blockIdx.x * 8 + wave;      // 2048 waves total
    int islice = w & 3;                 // 4 slices of the i loop
    int jtile  = (w >> 2) & 127;        // 128 j-tiles
    int b      = w >> 9;                // 4 batches

    bool lowhalf = lane < 16;
    int n  = lane & 15;
    int gj = jtile * 16 + n;
    const float4* jr = (const float4*)(ws + (size_t)(b * NP + gj) * 8);
    float4 jlo = jr[0];                 // x y z |p|^2
    float4 jhi = jr[1];                 // q rad 0 0
    float normj = jlo.w, qj = jhi.x, radj = jhi.y;
    int rowbase_j = rowbase_of(gj);

    // B operand: 4x16, rows K = x,y,z,0 striped across lane halves
    v2f Bop;
    Bop.x = lowhalf ? jlo.x : jlo.z;
    Bop.y = lowhalf ? jlo.y : 0.0f;

    float sx = 0.0f, sy = 0.0f, sz = 0.0f, sg = 0.0f;
    int rowadd = (lane >> 4) << 3;      // D layout: M = v + 8*(lane>=16)

    for (int ii = 0; ii < 32; ++ii) {
        int it = islice * 32 + ii;
        int ibase = b * NP + it * 16;

        if (ii < 31)
            __builtin_prefetch(ws + (size_t)(ibase + 16 + n) * 8, 0, 1);

        // A operand: 16x4, row m = lane&15, K split across lane halves.
        // Full b128 load, lane-half select done in registers (v_cndmask).
        float4 av = *(const float4*)(ws + (size_t)(ibase + n) * 8);
        v2f Aop;
        Aop.x = lowhalf ? av.x : av.z;
        Aop.y = lowhalf ? av.y : 0.0f;

        v8f acc = {0.f, 0.f, 0.f, 0.f, 0.f, 0.f, 0.f, 0.f};
        acc = __builtin_amdgcn_wmma_f32_16x16x4_f32(
            false, Aop, false, Bop, (short)0, acc, false, false);

#pragma unroll
        for (int v = 0; v < 8; ++v) {
            int row = v + rowadd;
            int gi  = it * 16 + row;
            const float* rp = ws + (size_t)(b * NP + gi) * 8;
            float4 lo = *(const float4*)rp;          // x y z |p|^2
            float2 hi = *(const float2*)(rp + 4);    // q rad

            float d2 = lo.w + normj - 2.0f * acc[v] + FEPS;

            int dmin = gi < gj ? gi : gj;
            int dmax = gi < gj ? gj : gi;
            bool valid = (dmax - dmin) > 2;      // excludes self, bonds, 1-3 pairs

            // analytic index into compacted nb list: lin(a,b) - 2a - 2
            int rb = (gi < gj) ? rowbase_of(gi) : rowbase_j;
            int k  = rb + dmax - 3 * dmin - 3;
            k = valid ? k : 0;
            int tid = nb_tid[k];
            float Ad = sDisp[tid];
            float Bh = sHbond[tid];

            float inv  = rsqrtf(d2);
            float inv2 = inv * inv;
            float inv3 = inv2 * inv;
            float sig  = hi.y + radj;
            float s2   = sig * sig * inv2;
            float sr6  = s2 * s2 * s2;
            float sr12 = sr6 * sr6;
            // f_j = g * (p_i - p_j)
            float gv = -Cc * hi.x * qj * inv3
                       - (12.0f * Ad * sr12 - 6.0f * Bh * sr6) * inv2;
            float g = valid ? gv : 0.0f;

            sx = fmaf(g, lo.x, sx);
            sy = fmaf(g, lo.y, sy);
            sz = fmaf(g, lo.z, sz);
            sg += g;
        }
    }

    // combine the two half-wave partials (rows 0-7 and 8-15 of each column)
    sx += __shfl_xor(sx, 16);
    sy += __shfl_xor(sy, 16);
    sz += __shfl_xor(sz, 16);
    sg += __shfl_xor(sg, 16);

    if (lane < 16) {
        V3 f = {sx - jlo.x * sg, sy - jlo.y * sg, sz - jlo.z * sg};
        atomicAdd3(out, b * NP + gj, f);
    }
}

// -------------------- bonds --------------------
__global__ void __launch_bounds__(256) k_bonds(const float* __restrict__ pos,
                                               const int* __restrict__ bond_idx,
                                               const int* __restrict__ bond_tid,
                                               const float* __restrict__ spring_k,
                                               const float* __restrict__ bond_d0,
                                               float* __restrict__ out, int nbond) {
    int t = blockIdx.x * 256 + threadIdx.x;
    if (t >= NBATCH * nbond) return;
    int b = t / nbond, e = t - b * nbond;
    int i0 = bond_idx[2 * e], i1 = bond_idx[2 * e + 1];
    int tid = bond_tid[e];
    V3 p0 = loadp(pos, b * NP + i0);
    V3 p1 = loadp(pos, b * NP + i1);
    V3 rb = v3sub(p0, p1);
    float d = sqrtf(v3dot(rb, rb) + FEPS);
    float c = -spring_k[tid] * (d - bond_d0[tid]) / d;
    V3 f0 = v3scale(rb, c);
    atomicAdd3(out, b * NP + i0, f0);
    atomicAdd3(out, b * NP + i1, v3scale(f0, -1.0f));
}

// -------------------- angles --------------------
__global__ void __launch_bounds__(256) k_angles(const float* __restrict__ pos,
                                                const int* __restrict__ angle_idx,
                                                const int* __restrict__ angle_tid,
                                                const float* __restrict__ angle_k,
                                                const float* __restrict__ angle_th0,
                                                float* __restrict__ out, int nang) {
    int t = blockIdx.x * 256 + threadIdx.x;
    if (t >= NBATCH * nang) return;
    int b = t / nang, e = t - b * nang;
    int i0 = angle_idx[3 * e], i1 = angle_idx[3 * e + 1], i2 = angle_idx[3 * e + 2];
    int tid = angle_tid[e];
    V3 p0 = loadp(pos, b * NP + i0);
    V3 p1 = loadp(pos, b * NP + i1);
    V3 p2 = loadp(pos, b * NP + i2);
    V3 u = v3sub(p0, p1), v = v3sub(p2, p1);
    float uu = v3dot(u, u), vv = v3dot(v, v), uv = v3dot(u, v);
    float s = sqrtf(uu * vv + FEPS);
    float invs = 1.0f / s;
    float c = uv * invs;
    const float lim = 1.0f - 1e-6f;
    float cl = fminf(fmaxf(c, -lim), lim);
    float theta = acosf(cl);
    float dEdth = angle_k[tid] * (theta - angle_th0[tid]);
    float dth_dc = -rsqrtf(1.0f - cl * cl);
    float pass = (fabsf(c) < lim) ? 1.0f : 0.0f;   // clip() zero-grad region
    float coef = dEdth * dth_dc * pass;
    float invs3 = invs * invs * invs;
    V3 dcdu = v3sub(v3scale(v, invs), v3scale(u, uv * vv * invs3));
    V3 dcdv = v3sub(v3scale(u, invs), v3scale(v, uv * uu * invs3));
    V3 g0 = v3scale(dcdu, coef);
    V3 g2 = v3scale(dcdv, coef);
    V3 g1 = v3scale(v3add(g0, g2), -1.0f);
    atomicAdd3(out, b * NP + i0, v3scale(g0, -1.0f));
    atomicAdd3(out, b * NP + i1, v3scale(g1, -1.0f));
    atomicAdd3(out, b * NP + i2, v3scale(g2, -1.0f));
}

// -------------------- dihedrals --------------------
__global__ void __launch_bounds__(256) k_dihedrals(const float* __restrict__ pos,
                                                   const int* __restrict__ dih_idx,
                                                   const int* __restrict__ dih_tid,
                                                   const float* __restrict__ dih_k,
                                                   const float* __restrict__ dih_phi0,
                                                   float* __restrict__ out, int ndih) {
    int t = blockIdx.x * 256 + threadIdx.x;
    if (t >= NBATCH * ndih) return;
    int b = t / ndih, e = t - b * ndih;
    int i0 = dih_idx[4 * e],     i1 = dih_idx[4 * e + 1];
    int i2 = dih_idx[4 * e + 2], i3 = dih_idx[4 * e + 3];
    int tid = dih_tid[e];
    V3 p0 = loadp(pos, b * NP + i0);
    V3 p1 = loadp(pos, b * NP + i1);
    V3 p2 = loadp(pos, b * NP + i2);
    V3 p3 = loadp(pos, b * NP + i3);
    V3 b1 = v3sub(p1, p0), b2 = v3sub(p2, p1), b3 = v3sub(p3, p2);
    V3 n1 = v3cross(b1, b2), n2 = v3cross(b2, b3);
    float b2sq  = v3dot(b2, b2) + FEPS;
    float b2len = sqrtf(b2sq);
    float n1sq  = v3dot(n1, n1) + FEPS;
    float n2sq  = v3dot(n2, n2) + FEPS;
    V3 m1 = v3cross(n1, v3scale(b2, 1.0f / b2len));
    float phi = atan2f(v3dot(m1, n2), v3dot(n1, n2));
    float dE = dih_k[tid] * (phi - dih_phi0[tid]);
    V3 dp0 = v3scale(n1, -b2len / n1sq);
    V3 dp3 = v3scale(n2,  b2len / n2sq);
    float c1 = v3dot(b1, b2) / b2sq;
    float c3 = v3dot(b3, b2) / b2sq;
    V3 dp1 = v3sub(v3scale(dp0, c1 - 1.0f), v3scale(dp3, c3));
    V3 dp2 = v3sub(v3scale(dp3, c3 - 1.0f), v3scale(dp0, c1));
    atomicAdd3(out, b * NP + i0, v3scale(dp0, -dE));
    atomicAdd3(out, b * NP + i1, v3scale(dp1, -dE));
    atomicAdd3(out, b * NP + i2, v3scale(dp2, -dE));
    atomicAdd3(out, b * NP + i3, v3scale(dp3, -dE));
}

extern "C" void kernel_launch(void* const* d_in, const int* in_sizes, int n_in,
                              void* d_out, int out_size, void* d_ws, size_t ws_size,
                              hipStream_t stream) {
    const float* pos        = (const float*)d_in[0];
    const float* bead_radii = (const float*)d_in[1];
    const float* q_vals     = (const float*)d_in[2];
    const float* spring_k   = (const float*)d_in[3];
    const float* bond_d0    = (const float*)d_in[4];
    const float* angle_k    = (const float*)d_in[5];
    const float* angle_th0  = (const float*)d_in[6];
    const float* dih_k      = (const float*)d_in[7];
    const float* dih_phi0   = (const float*)d_in[8];
    const float* disp_k     = (const float*)d_in[9];
    const float* hbond_k    = (const float*)d_in[10];
    const float* e_r        = (const float*)d_in[11];
    const float* f_0        = (const float*)d_in[12];
    const int*   bead_types = (const int*)d_in[13];
    const int*   charge_idx = (const int*)d_in[14];
    const int*   bond_idx   = (const int*)d_in[15];
    const int*   bond_tid   = (const int*)d_in[16];
    const int*   angle_idx  = (const int*)d_in[17];
    const int*   angle_tid  = (const int*)d_in[18];
    const int*   dih_idx    = (const int*)d_in[19];
    const int*   dih_tid    = (const int*)d_in[20];
    const int*   nb_tid     = (const int*)d_in[23];

    float* out = (float*)d_out;
    float* ws  = (float*)d_ws;

    int nbond = in_sizes[16];   // 2047
    int nang  = in_sizes[18];   // 2046
    int ndih  = in_sizes[20];   // 512

    k_zero<<<(out_size + 255) / 256, 256, 0, stream>>>(out, out_size);
    k_prep<<<(NBATCH * NP + 255) / 256, 256, 0, stream>>>(
        pos, q_vals, charge_idx, bead_radii, bead_types, ws);
    // 4 batches * 128 j-tiles * 4 i-slices = 2048 waves = 256 blocks of 8 waves
    k_nonbond<<<256, 256, 0, stream>>>(ws, nb_tid, disp_k, hbond_k, f_0, e_r, out);
    k_bonds<<<(NBATCH * nbond + 255) / 256, 256, 0, stream>>>(
        pos, bond_idx, bond_tid, spring_k, bond_d0, out, nbond);
    k_angles<<<(NBATCH * nang + 255) / 256, 256, 0, stream>>>(
        pos, angle_idx, angle_tid, angle_k, angle_th0, out, nang);
    k_dihedrals<<<(NBATCH * ndih + 255) / 256, 256, 0, stream>>>(
        pos, dih_idx, dih_tid, dih_k, dih_phi0, out, ndih);
}